// GCNLayer_49323404427940
// MI455X (gfx1250) — compile-verified
//
#include <hip/hip_runtime.h>
#include <math.h>

#define N_NODES 50000
#define N_EDGES 800000
#define N_GRAPH 250
#define T_SL    8
#define C_IN    32
#define C_OUT   64
#define NODES_PER_GRAPH (N_NODES / N_GRAPH)   // 200

typedef __attribute__((ext_vector_type(2))) float v2f;
typedef __attribute__((ext_vector_type(8))) float v8f;

// ---------------------------------------------------------------------------
// FP32 WMMA GEMM: out[M x 64] = A[M x K] @ W[K x 64]   (M = 50000, 16-row tiles)
// One wave computes one 16x16 tile via V_WMMA_F32_16X16X4_F32; 4 waves/block
// cover the 64 output channels. Optional fused ReLU on the A operand.
// A fragment (16x4 fp32, ISA 7.12.2): lanes 0-15 hold K=k0,k0+1 of row M=lane;
// lanes 16-31 hold K=k0+2,k0+3 of row M=lane-16. B fragment mirrors with N=lane.
// D: VGPR r = row (r + 8*half), col = lane&15.
// ---------------------------------------------------------------------------
__global__ __launch_bounds__(128) void gemm_wmma_kernel(
    const float* __restrict__ A, int rowStride, int colStride,
    const float* __restrict__ W, int K, int applyRelu,
    float* __restrict__ out)
{
    const int lane = threadIdx.x & 31;
    const int wave = threadIdx.x >> 5;        // 0..3
    const int m0   = blockIdx.x * 16;         // node tile (50000/16 = 3125 exact)
    const int n0   = wave * 16;               // cout tile
    const int half = lane >> 4;               // K-half select
    const int l    = lane & 15;

    const float* aRow = A + (size_t)(m0 + l) * rowStride;
    const float* wCol = W + n0 + l;

    v8f acc = {};
    for (int k0 = 0; k0 < K; k0 += 4) {
        const int ka = k0 + 2 * half;
        float a0 = aRow[(size_t)ka * colStride];
        float a1 = aRow[(size_t)(ka + 1) * colStride];
        if (applyRelu) { a0 = fmaxf(a0, 0.0f); a1 = fmaxf(a1, 0.0f); }
        v2f a; a.x = a0; a.y = a1;
        v2f b; b.x = wCol[(size_t)ka * C_OUT]; b.y = wCol[(size_t)(ka + 1) * C_OUT];
        acc = __builtin_amdgcn_wmma_f32_16x16x4_f32(
            /*neg_a=*/false, a, /*neg_b=*/false, b,
            /*c_mod=*/(short)0, acc, /*reuse_a=*/false, /*reuse_b=*/false);
    }

    float* orow = out + (size_t)(m0 + 8 * half) * C_OUT + n0 + l;
#pragma unroll
    for (int r = 0; r < 8; ++r)
        orow[(size_t)r * C_OUT] = acc[r];
}

// ---------------------------------------------------------------------------
// Setup kernels: degree (with self-loop), dinv, edge norms
// ---------------------------------------------------------------------------
__global__ void init_deg_kernel(float* __restrict__ deg, int n)
{
    int i = blockIdx.x * blockDim.x + threadIdx.x;
    if (i < n) deg[i] = 1.0f;                 // self-loop contribution
}

__global__ void count_deg_kernel(const int* __restrict__ dst, float* __restrict__ deg, int e)
{
    int i = blockIdx.x * blockDim.x + threadIdx.x;
    if (i < e) atomicAdd(&deg[dst[i]], 1.0f);
}

__global__ void dinv_kernel(const float* __restrict__ deg, float* __restrict__ dinv,
                            float* __restrict__ norm, int n)
{
    int i = blockIdx.x * blockDim.x + threadIdx.x;
    if (i < n) {
        float di = rsqrtf(deg[i]);
        dinv[i] = di;
        norm[N_EDGES + i] = di * di;          // self-loop norm
    }
}

__global__ void edge_norm_kernel(const int* __restrict__ src, const int* __restrict__ dst,
                                 const float* __restrict__ dinv, float* __restrict__ norm, int e)
{
    int i = blockIdx.x * blockDim.x + threadIdx.x;
    if (i < e) norm[i] = dinv[src[i]] * dinv[dst[i]];
}

// ---------------------------------------------------------------------------
// Per-layer kernels
// ---------------------------------------------------------------------------
__global__ void bias_init_kernel(float* __restrict__ h, const float* __restrict__ bias, int total)
{
    int i = blockIdx.x * blockDim.x + threadIdx.x;
    if (i < total) h[i] = bias[i & (C_OUT - 1)];
}

// 16 threads per edge, each handling 4 channels (float4). L2-resident atomics.
__global__ void scatter_kernel(const float* __restrict__ hW,
                               const int* __restrict__ src, const int* __restrict__ dst,
                               const float* __restrict__ norm,
                               float* __restrict__ hA, int nTot)
{
    int idx = blockIdx.x * blockDim.x + threadIdx.x;
    int e = idx >> 4;
    if (e >= nTot) return;
    int c4 = (idx & 15) << 2;
    int s, d;
    if (e < N_EDGES) { s = src[e]; d = dst[e]; }
    else             { s = d = e - N_EDGES; }           // self-loop
    float w = norm[e];
    const float4 v = *reinterpret_cast<const float4*>(hW + (size_t)s * C_OUT + c4);
    float* o = hA + (size_t)d * C_OUT + c4;
    atomicAdd(o + 0, w * v.x);
    atomicAdd(o + 1, w * v.y);
    atomicAdd(o + 2, w * v.z);
    atomicAdd(o + 3, w * v.w);
}

// One block per graph (200 contiguous nodes), one thread per channel.
// Applies ReLU on the fly, accumulates max + mean into out[b, :, t].
__global__ __launch_bounds__(C_OUT) void pool_kernel(const float* __restrict__ h,
                                                     float* __restrict__ out, int t)
{
    int ch = threadIdx.x;                                // 0..63
    int b  = blockIdx.x;                                 // 0..249
    const float* p = h + (size_t)b * NODES_PER_GRAPH * C_OUT + ch;
    float mx = -INFINITY, sm = 0.0f;
    for (int n = 0; n < NODES_PER_GRAPH; ++n) {
        float v = fmaxf(p[(size_t)n * C_OUT], 0.0f);     // relu
        mx = fmaxf(mx, v);
        sm += v;
    }
    float* ob = out + (size_t)b * (2 * C_OUT * T_SL);
    ob[ch * T_SL + t]           += mx;
    ob[(C_OUT + ch) * T_SL + t] += sm * (1.0f / NODES_PER_GRAPH);
}

__global__ void zero_kernel(float* __restrict__ p, int n)
{
    int i = blockIdx.x * blockDim.x + threadIdx.x;
    if (i < n) p[i] = 0.0f;
}

// ---------------------------------------------------------------------------
// Host orchestration
// ---------------------------------------------------------------------------
extern "C" void kernel_launch(void* const* d_in, const int* in_sizes, int n_in,
                              void* d_out, int out_size, void* d_ws, size_t ws_size,
                              hipStream_t stream)
{
    const float* x  = (const float*)d_in[0];   // [N, CIN, T]
    const int*   ei = (const int*)d_in[1];     // [2, E]
    const float* W1 = (const float*)d_in[3];
    const float* b1 = (const float*)d_in[4];
    const float* W2 = (const float*)d_in[5];
    const float* b2 = (const float*)d_in[6];
    const float* W3 = (const float*)d_in[7];
    const float* b3 = (const float*)d_in[8];
    float* out = (float*)d_out;

    const int* src = ei;                       // edge_index[0]
    const int* dst = ei + N_EDGES;             // edge_index[1]

    // Workspace layout (floats)
    float* ws   = (float*)d_ws;
    float* deg  = ws;                                  // N
    float* dinv = deg + N_NODES;                       // N
    float* norm = dinv + N_NODES;                      // E + N
    float* hW   = norm + (N_EDGES + N_NODES);          // N * 64 (gemm result)
    float* hA   = hW + (size_t)N_NODES * C_OUT;        // N * 64 (aggregated h)

    const int nTot = N_EDGES + N_NODES;

    // ---- setup: degree / dinv / norm ----
    init_deg_kernel<<<(N_NODES + 255) / 256, 256, 0, stream>>>(deg, N_NODES);
    count_deg_kernel<<<(N_EDGES + 255) / 256, 256, 0, stream>>>(dst, deg, N_EDGES);
    dinv_kernel<<<(N_NODES + 255) / 256, 256, 0, stream>>>(deg, dinv, norm, N_NODES);
    edge_norm_kernel<<<(N_EDGES + 255) / 256, 256, 0, stream>>>(src, dst, dinv, norm, N_EDGES);

    // ---- zero output (accumulated across layers/slices) ----
    zero_kernel<<<(N_GRAPH * 2 * C_OUT * T_SL + 255) / 256, 256, 0, stream>>>(
        out, N_GRAPH * 2 * C_OUT * T_SL);

    const dim3 gemmGrid(N_NODES / 16);        // 3125
    const dim3 gemmBlock(128);                // 4 waves -> 4 x 16 couts
    const int hTotal = N_NODES * C_OUT;
    const int scatThreads = nTot * 16;

    for (int t = 0; t < T_SL; ++t) {
        // layer 1: A = x[:, :, t]  (row stride CIN*T, col stride T)
        gemm_wmma_kernel<<<gemmGrid, gemmBlock, 0, stream>>>(
            x + t, C_IN * T_SL, T_SL, W1, C_IN, /*relu=*/0, hW);
        bias_init_kernel<<<(hTotal + 255) / 256, 256, 0, stream>>>(hA, b1, hTotal);
        scatter_kernel<<<(scatThreads + 255) / 256, 256, 0, stream>>>(hW, src, dst, norm, hA, nTot);
        pool_kernel<<<N_GRAPH, C_OUT, 0, stream>>>(hA, out, t);

        // layer 2: A = relu(hA)
        gemm_wmma_kernel<<<gemmGrid, gemmBlock, 0, stream>>>(
            hA, C_OUT, 1, W2, C_OUT, /*relu=*/1, hW);
        bias_init_kernel<<<(hTotal + 255) / 256, 256, 0, stream>>>(hA, b2, hTotal);
        scatter_kernel<<<(scatThreads + 255) / 256, 256, 0, stream>>>(hW, src, dst, norm, hA, nTot);
        pool_kernel<<<N_GRAPH, C_OUT, 0, stream>>>(hA, out, t);

        // layer 3: A = relu(hA)
        gemm_wmma_kernel<<<gemmGrid, gemmBlock, 0, stream>>>(
            hA, C_OUT, 1, W3, C_OUT, /*relu=*/1, hW);
        bias_init_kernel<<<(hTotal + 255) / 256, 256, 0, stream>>>(hA, b3, hTotal);
        scatter_kernel<<<(scatThreads + 255) / 256, 256, 0, stream>>>(hW, src, dst, norm, hA, nTot);
        pool_kernel<<<N_GRAPH, C_OUT, 0, stream>>>(hA, out, t);
    }
}